// GCNEncoder_206158430595
// MI455X (gfx1250) — compile-verified
//
#include <hip/hip_runtime.h>
#include <math.h>

// ---------------------------------------------------------------------------
// GCN encoder for MI455X (gfx1250, wave32).
//   h1 = relu(LN(Dinvhalf (A+I) Dinvhalf (x@W1) + b1))
//   h2 =        Dinvhalf (A+I) Dinvhalf (h1@W2) + b2
//   out = graph-mean-pool(h2)
// GEMMs use V_WMMA_F32_16X16X4_F32 (fp32 matrix pipe, matches reference
// precision). Scatter uses native global_atomic_add_f32 (unsafeAtomicAdd).
// A-strip staging uses gfx1250 async global->LDS loads when available.
// ---------------------------------------------------------------------------

#define DIM 128
#define LDS_STRIDE 132   // 132 % 64 == 4 -> conflict-free column reads; 528B row (16-aligned)

typedef __attribute__((ext_vector_type(2))) float v2f;
typedef __attribute__((ext_vector_type(8))) float v8f;
typedef __attribute__((ext_vector_type(4))) int   v4i;

#if __has_builtin(__builtin_amdgcn_global_load_async_to_lds_b128) && \
    __has_builtin(__builtin_amdgcn_s_wait_asynccnt)
#define HAVE_ASYNC_LDS 1
typedef v4i __attribute__((address_space(1)))* g_v4i_ptr;   // global int4*
typedef v4i __attribute__((address_space(3)))* l_v4i_ptr;   // LDS int4*
#endif

// ---------------- init: zero accumulators, deg = 1 (self loop) -------------
__global__ __launch_bounds__(256) void k_init(float* __restrict__ agg,
                                              float* __restrict__ deg,
                                              float* __restrict__ pool,
                                              float* __restrict__ cnt,
                                              int n, int nd, int pd, int g) {
    int i = blockIdx.x * 256 + threadIdx.x;
    if (i < nd) agg[i] = 0.0f;
    if (i < n)  deg[i] = 1.0f;          // +1 self loop
    if (i < pd) pool[i] = 0.0f;
    if (i < g)  cnt[i]  = 0.0f;
}

__global__ __launch_bounds__(256) void k_zero(float* __restrict__ p, int n) {
    int i = blockIdx.x * 256 + threadIdx.x;
    if (i < n) p[i] = 0.0f;
}

// ---------------- degree + rsqrt -------------------------------------------
__global__ __launch_bounds__(256) void k_degree(const int* __restrict__ dst,
                                                float* __restrict__ deg, int e) {
    int i = blockIdx.x * 256 + threadIdx.x;
    if (i < e) unsafeAtomicAdd(&deg[dst[i]], 1.0f);
}

__global__ __launch_bounds__(256) void k_rsqrt(float* __restrict__ deg, int n) {
    int i = blockIdx.x * 256 + threadIdx.x;
    if (i < n) deg[i] = rsqrtf(deg[i]);
}

// ---------------- fp32 WMMA GEMM: Out[M,128] = A[M,128] @ W[128,128] -------
// One block = 16-row strip of A (staged in LDS); 8 waves cover 8 x 16 cols.
__global__ __launch_bounds__(256) void k_gemm_wmma(const float* __restrict__ A,
                                                   const float* __restrict__ W,
                                                   float* __restrict__ Out,
                                                   int M) {
    __shared__ __align__(16) float lds[16 * LDS_STRIDE];
    const int m0  = blockIdx.x * 16;
    const int tid = threadIdx.x;

    // stage 16x128 A strip (512 float4 chunks; row stride 528B keeps 16-align)
#ifdef HAVE_ASYNC_LDS
    for (int j = tid; j < 16 * (DIM / 4); j += 256) {
        int r = j >> 5, c4 = j & 31;
        int gr = m0 + r; if (gr >= M) gr = M - 1;   // clamp (M%16==0 normally)
        __builtin_amdgcn_global_load_async_to_lds_b128(
            (g_v4i_ptr)(A + (long)gr * DIM + c4 * 4),
            (l_v4i_ptr)&lds[r * LDS_STRIDE + c4 * 4],
            /*offset=*/0, /*cpol=*/0);
    }
    __builtin_amdgcn_s_wait_asynccnt(0);
    __syncthreads();
#else
    for (int j = tid; j < 16 * (DIM / 4); j += 256) {
        int r = j >> 5, c4 = j & 31;
        int gr = m0 + r; if (gr >= M) gr = M - 1;
        const float4 v = *(const float4*)(A + (long)gr * DIM + c4 * 4);
        *(float4*)&lds[r * LDS_STRIDE + c4 * 4] = v;
    }
    __syncthreads();
#endif

    const int wave = tid >> 5;          // 0..7 -> column tile
    const int lane = tid & 31;
    const int half = lane >> 4;         // 0: K pair (k0,k0+1), 1: (k0+2,k0+3)
    const int l16  = lane & 15;
    const int n0   = wave * 16;

    v8f acc = {};
    #pragma unroll
    for (int k0 = 0; k0 < DIM; k0 += 4) {
        const int lo = k0 + 2 * half;
        v2f a, b;
        // A 16x4 layout: lanes 0-15 rows, VGPR0/1 = K=lo,lo+1 (hi half K+2)
        a.x = lds[l16 * LDS_STRIDE + lo];
        a.y = lds[l16 * LDS_STRIDE + lo + 1];
        // B 4x16 layout: row striped across lanes within a VGPR
        b.x = W[(lo    ) * DIM + n0 + l16];
        b.y = W[(lo + 1) * DIM + n0 + l16];
        acc = __builtin_amdgcn_wmma_f32_16x16x4_f32(
            /*neg_a=*/false, a, /*neg_b=*/false, b,
            /*c_mod=*/(short)0, acc, /*reuse_a=*/false, /*reuse_b=*/false);
    }

    // C/D layout: VGPR r -> row r (lanes 0-15) / row r+8 (lanes 16-31)
    #pragma unroll
    for (int r = 0; r < 8; ++r) {
        int row = m0 + r + 8 * half;
        if (row < M) Out[(long)row * DIM + n0 + l16] = acc[r];
    }
}

// ---------------- edge scatter: agg[dst] += coef * xw[src] -----------------
// One wave per edge; lane handles 4 contiguous floats (float4 gather,
// 4 native f32 atomics). ~L2-resident gathers, atomic-throughput bound.
__global__ __launch_bounds__(256) void k_scatter(const float* __restrict__ xw,
                                                 const int* __restrict__ src,
                                                 const int* __restrict__ dst,
                                                 const float* __restrict__ dinv,
                                                 float* __restrict__ agg, int e) {
    int edge = blockIdx.x * 8 + (threadIdx.x >> 5);
    if (edge >= e) return;
    int lane = threadIdx.x & 31;
    int s = src[edge], d = dst[edge];
    float coef = dinv[s] * dinv[d];
    float4 v = ((const float4*)(xw + (long)s * DIM))[lane];
    float* a = agg + (long)d * DIM + lane * 4;
    unsafeAtomicAdd(a + 0, v.x * coef);
    unsafeAtomicAdd(a + 1, v.y * coef);
    unsafeAtomicAdd(a + 2, v.z * coef);
    unsafeAtomicAdd(a + 3, v.w * coef);
}

// ---------------- layer-1 epilogue: self-loop + bias + LN + ReLU -----------
// One wave per node; 128 features = 32 lanes x float4; shfl_xor reductions.
__global__ __launch_bounds__(256) void k_ep1(const float* __restrict__ xw,
                                             float* __restrict__ agg,  // in/out
                                             const float* __restrict__ dinv,
                                             const float* __restrict__ bias,
                                             const float* __restrict__ gamma,
                                             const float* __restrict__ beta,
                                             int n) {
    int node = blockIdx.x * 8 + (threadIdx.x >> 5);
    if (node >= n) return;
    int lane = threadIdx.x & 31;
    float di = dinv[node], d2 = di * di;
    float4 h  = ((const float4*)(agg + (long)node * DIM))[lane];
    float4 xv = ((const float4*)(xw  + (long)node * DIM))[lane];
    float4 bb = ((const float4*)bias)[lane];
    h.x += xv.x * d2 + bb.x;  h.y += xv.y * d2 + bb.y;
    h.z += xv.z * d2 + bb.z;  h.w += xv.w * d2 + bb.w;

    float s = h.x + h.y + h.z + h.w;
    #pragma unroll
    for (int off = 16; off > 0; off >>= 1) s += __shfl_xor(s, off);
    float mu = s * (1.0f / 128.0f);

    float4 dv = {h.x - mu, h.y - mu, h.z - mu, h.w - mu};
    float q = dv.x * dv.x + dv.y * dv.y + dv.z * dv.z + dv.w * dv.w;
    #pragma unroll
    for (int off = 16; off > 0; off >>= 1) q += __shfl_xor(q, off);
    float inv = rsqrtf(q * (1.0f / 128.0f) + 1e-5f);

    float4 gm = ((const float4*)gamma)[lane];
    float4 bt = ((const float4*)beta)[lane];
    float4 o;
    o.x = fmaxf(dv.x * inv * gm.x + bt.x, 0.0f);
    o.y = fmaxf(dv.y * inv * gm.y + bt.y, 0.0f);
    o.z = fmaxf(dv.z * inv * gm.z + bt.z, 0.0f);
    o.w = fmaxf(dv.w * inv * gm.w + bt.w, 0.0f);
    ((float4*)(agg + (long)node * DIM))[lane] = o;
}

// ---------------- layer-2 epilogue: self-loop + bias + pool atomics --------
__global__ __launch_bounds__(256) void k_ep2(const float* __restrict__ xw,
                                             const float* __restrict__ agg,
                                             const float* __restrict__ dinv,
                                             const float* __restrict__ bias,
                                             const int* __restrict__ batch,
                                             float* __restrict__ pool,
                                             float* __restrict__ cnt, int n) {
    int node = blockIdx.x * 8 + (threadIdx.x >> 5);
    if (node >= n) return;
    int lane = threadIdx.x & 31;
    float di = dinv[node], d2 = di * di;
    float4 h  = ((const float4*)(agg + (long)node * DIM))[lane];
    float4 xv = ((const float4*)(xw  + (long)node * DIM))[lane];
    float4 bb = ((const float4*)bias)[lane];
    h.x += xv.x * d2 + bb.x;  h.y += xv.y * d2 + bb.y;
    h.z += xv.z * d2 + bb.z;  h.w += xv.w * d2 + bb.w;

    int g = batch[node];
    float* p = pool + (long)g * DIM + lane * 4;
    unsafeAtomicAdd(p + 0, h.x);
    unsafeAtomicAdd(p + 1, h.y);
    unsafeAtomicAdd(p + 2, h.z);
    unsafeAtomicAdd(p + 3, h.w);
    if (lane == 0) unsafeAtomicAdd(&cnt[g], 1.0f);
}

// ---------------- final divide ---------------------------------------------
__global__ __launch_bounds__(256) void k_pool_div(const float* __restrict__ pool,
                                                  const float* __restrict__ cnt,
                                                  float* __restrict__ out, int gd) {
    int i = blockIdx.x * 256 + threadIdx.x;
    if (i < gd) out[i] = pool[i] / fmaxf(cnt[i >> 7], 1.0f);
}

// ---------------------------------------------------------------------------
extern "C" void kernel_launch(void* const* d_in, const int* in_sizes, int n_in,
                              void* d_out, int out_size, void* d_ws, size_t ws_size,
                              hipStream_t stream) {
    const float* x     = (const float*)d_in[0];
    const int*   src   = (const int*)  d_in[1];
    const int*   dst   = (const int*)  d_in[2];
    const int*   batch = (const int*)  d_in[3];
    const float* W1    = (const float*)d_in[4];
    const float* b1    = (const float*)d_in[5];
    const float* gamma = (const float*)d_in[6];
    const float* beta  = (const float*)d_in[7];
    const float* W2    = (const float*)d_in[8];
    const float* b2    = (const float*)d_in[9];
    float* out = (float*)d_out;

    const int N = in_sizes[0] / DIM;     // 50000
    const int E = in_sizes[1];           // 800000
    const int G = out_size / DIM;        // 64
    const long ND = (long)N * DIM;

    float* bufA = (float*)d_ws;          // xw  (GEMM output)
    float* bufB = bufA + ND;             // agg / h1
    float* dinv = bufB + ND;             // degree -> rsqrt(deg)
    float* pool = dinv + N;              // G x DIM
    float* cnt  = pool + (long)G * DIM;  // G

    const int threads = 256;
    const int gInit    = (int)((ND + threads - 1) / threads);
    const int gEdge    = (E + threads - 1) / threads;
    const int gNode    = (N + threads - 1) / threads;
    const int gGemm    = (N + 15) / 16;
    const int gWaveE   = (E + 7) / 8;    // 1 wave / edge
    const int gWaveN   = (N + 7) / 8;    // 1 wave / node
    const int gPool    = (G * DIM + threads - 1) / threads;

    // degree / normalization
    k_init  <<<gInit, threads, 0, stream>>>(bufB, dinv, pool, cnt,
                                            N, (int)ND, G * DIM, G);
    k_degree<<<gEdge, threads, 0, stream>>>(dst, dinv, E);
    k_rsqrt <<<gNode, threads, 0, stream>>>(dinv, N);

    // ---- layer 1 ----
    k_gemm_wmma<<<gGemm, threads, 0, stream>>>(x, W1, bufA, N);
    k_scatter  <<<gWaveE, threads, 0, stream>>>(bufA, src, dst, dinv, bufB, E);
    k_ep1      <<<gWaveN, threads, 0, stream>>>(bufA, bufB, dinv, b1, gamma, beta, N);

    // ---- layer 2 ----
    k_gemm_wmma<<<gGemm, threads, 0, stream>>>(bufB, W2, bufA, N);
    k_zero     <<<gInit, threads, 0, stream>>>(bufB, (int)ND);
    k_scatter  <<<gWaveE, threads, 0, stream>>>(bufA, src, dst, dinv, bufB, E);
    k_ep2      <<<gWaveN, threads, 0, stream>>>(bufA, bufB, dinv, b2, batch, pool, cnt, N);

    // ---- pool ----
    k_pool_div <<<gPool, threads, 0, stream>>>(pool, cnt, out, G * DIM);
}